// Embed_38766374814290
// MI455X (gfx1250) — compile-verified
//
#include <hip/hip_runtime.h>

// ---------------------------------------------------------------------------
// out[i,j,l,e] = base[mi][e] + ds[i,j,l] * slope[mi][e]
//   mi    = (j < traj_len[i])
//   ds    = mi ? mat2[traj_loc[i,j]-1, l] : 0
//   slope = (emb_su-emb_sl)/SU + (emb_tu-emb_tl)/TU   (SL = TL = 0)
//   base  = emb_sl + emb_tl
// Rank-1 GEMM per (i,j): D(16x16) = A(16x4:[ds|0|0|0]) x B(4x16:[slope;0;0;0]) + C(base)
// Store-bandwidth bound: 128 MiB out @ 23.3 TB/s ~= 5.6 us floor.
// ---------------------------------------------------------------------------

typedef __attribute__((ext_vector_type(2))) float v2f;
typedef __attribute__((ext_vector_type(8))) float v8f;

#define N_TRAJ   32
#define M_TRAJ   128
#define L_MAX_C  128
#define EMB_C    64
#define SU_C     10000.0f
#define TU_C     86400.0f

__global__ __launch_bounds__(256) void embed_interval_wmma(
    const int*   __restrict__ traj_loc,  // [N, M]   (1-indexed)
    const float* __restrict__ mat2,      // [N_LOC, L_MAX]
    const int*   __restrict__ traj_len,  // [N]
    const float* __restrict__ emb_sl,    // [2, EMB]
    const float* __restrict__ emb_su,    // [2, EMB]
    const float* __restrict__ emb_tl,    // [2, EMB]
    const float* __restrict__ emb_tu,    // [2, EMB]
    float*       __restrict__ out)       // [N, M, L_MAX, EMB]
{
    const int p      = blockIdx.x;        // (i,j) pair index, 0..N*M-1
    const int i      = p >> 7;            // / M_TRAJ
    const int j      = p & (M_TRAJ - 1);
    const int tid    = threadIdx.x;
    const int wave   = tid >> 5;          // 8 waves: wave w owns L rows [16w,16w+16)
    const int lane   = tid & 31;
    const int half   = lane >> 4;         // 0 => lanes 0-15, 1 => lanes 16-31
    const int lane16 = lane & 15;

    const int mi  = (j < traj_len[i]) ? 1 : 0;
    const int loc = traj_loc[p] - 1;
    const int l0  = wave * 16;

    // --- A matrix (16x4 f32, 2 VGPRs): K=0 column = ds, K=1..3 = 0 -----------
    // Layout: lanes 0-15 carry {K=0,K=1} for row M=lane; lanes 16-31 carry {K=2,K=3}.
    float dsv = 0.0f;
    if (mi && half == 0) {
        dsv = mat2[(size_t)loc * L_MAX_C + l0 + lane16];
    }
    v2f a;
    a.x = dsv;    // K=0 (lanes 0-15) / K=2 (lanes 16-31, forced 0 above)
    a.y = 0.0f;   // K=1 / K=3

    float* outp = out + (size_t)p * (L_MAX_C * EMB_C);

    // --- 4 E-tiles of 16 columns each -----------------------------------------
    #pragma unroll
    for (int et = 0; et < 4; ++et) {
        const int e0 = et * 16;
        const int e  = e0 + lane16;

        const float sl = emb_sl[mi * EMB_C + e];
        const float su = emb_su[mi * EMB_C + e];
        const float tl = emb_tl[mi * EMB_C + e];
        const float tu = emb_tu[mi * EMB_C + e];

        const float slope = (su - sl) * (1.0f / SU_C) + (tu - tl) * (1.0f / TU_C);
        const float basev = sl + tl;

        // B matrix (4x16 f32, 2 VGPRs): row K=0 = slope, K=1..3 = 0.
        v2f b;
        b.x = (half == 0) ? slope : 0.0f;
        b.y = 0.0f;

        // C: base[e] broadcast across all rows (VGPR r, lane l -> M=r+8*half, N=lane16)
        v8f c;
        #pragma unroll
        for (int r = 0; r < 8; ++r) c[r] = basev;

        // D = A x B + C   (exact f32 rank-1 update on the matrix pipe)
        v8f d = __builtin_amdgcn_wmma_f32_16x16x4_f32(
            /*neg_a=*/false, a, /*neg_b=*/false, b,
            /*c_mod=*/(short)0, c, /*reuse_a=*/false, /*reuse_b=*/false);

        // Store D: VGPR r holds rows (l0+r) and (l0+r+8); lanes give 64B-contig cols.
        // Write-once 128 MiB stream -> non-temporal hint.
        #pragma unroll
        for (int r = 0; r < 8; ++r) {
            const int row = l0 + r + 8 * half;
            __builtin_nontemporal_store(d[r], outp + (size_t)row * EMB_C + e0 + lane16);
        }
    }
}

extern "C" void kernel_launch(void* const* d_in, const int* in_sizes, int n_in,
                              void* d_out, int out_size, void* d_ws, size_t ws_size,
                              hipStream_t stream) {
    // setup_inputs() order:
    // 0: traj_loc [N,M] int, 1: mat2 [N_LOC,L_MAX] f32, 2: vec (unused),
    // 3: traj_len [N] int, 4: l_max scalar (unused, ==128),
    // 5..8: emb_sl_w, emb_su_w, emb_tl_w, emb_tu_w [2,EMB] f32
    const int*   traj_loc = (const int*)  d_in[0];
    const float* mat2     = (const float*)d_in[1];
    const int*   traj_len = (const int*)  d_in[3];
    const float* emb_sl   = (const float*)d_in[5];
    const float* emb_su   = (const float*)d_in[6];
    const float* emb_tl   = (const float*)d_in[7];
    const float* emb_tu   = (const float*)d_in[8];
    float*       out      = (float*)d_out;

    const int n_pairs = N_TRAJ * M_TRAJ;  // 4096 workgroups, 8 wave32s each
    embed_interval_wmma<<<n_pairs, 256, 0, stream>>>(
        traj_loc, mat2, traj_len, emb_sl, emb_su, emb_tl, emb_tu, out);
}